// Decoder_14078902796966
// MI455X (gfx1250) — compile-verified
//
#include <hip/hip_runtime.h>
#include <hip/hip_bf16.h>
#include <math.h>

// Problem dims
#define B_  16
#define T_  64
#define H_  512
#define D_  512
#define V_  50257
#define G_  2048   // 4*H
#define K2_ 1024   // 2*H

typedef __attribute__((ext_vector_type(2))) float v2f;
typedef __attribute__((ext_vector_type(8))) float v8f;
typedef __attribute__((ext_vector_type(4))) int   v4i;

#ifndef __has_builtin
#define __has_builtin(x) 0
#endif

#if __has_builtin(__builtin_amdgcn_global_load_async_to_lds_b128) && \
    __has_builtin(__builtin_amdgcn_s_wait_asynccnt)
#define USE_ASYNC_COPY 1
#else
#define USE_ASYNC_COPY 0
#endif

#define AS1_ __attribute__((address_space(1)))
#define AS3_ __attribute__((address_space(3)))

// 16-byte global -> LDS copy. Async (ASYNCcnt-tracked, no VGPR round trip) when
// the toolchain exposes the gfx1250 builtin; plain ds_store fallback otherwise.
// Builtin signature (from hipcc diagnostic): (AS1 int4*, AS3 int4*, Imm, Imm).
__device__ __forceinline__ void async_copy16(const float* gsrc, float* ldst) {
#if USE_ASYNC_COPY
  __builtin_amdgcn_global_load_async_to_lds_b128((AS1_ v4i*)gsrc, (AS3_ v4i*)ldst,
                                                 /*imm_offset=*/0, /*cpol=*/0);
#else
  *(float4*)ldst = *(const float4*)gsrc;
#endif
}

#if USE_ASYNC_COPY
#define ASYNC_WAIT(n) __builtin_amdgcn_s_wait_asynccnt(n)
#else
#define ASYNC_WAIT(n)
#endif

#if __has_builtin(__builtin_amdgcn_sched_barrier)
#define SCHED_FENCE() __builtin_amdgcn_sched_barrier(0)
#else
#define SCHED_FENCE()
#endif

// V_WMMA_F32_16X16X4_F32: D(16x16 f32) = A(16x4 f32) * B(4x16 f32) + C
__device__ __forceinline__ v8f wmma_f32_k4(v2f a, v2f b, v8f c) {
  return __builtin_amdgcn_wmma_f32_16x16x4_f32(
      /*neg_a=*/false, a, /*neg_b=*/false, b,
      /*c_mod=*/(short)0, c, /*reuse_a=*/false, /*reuse_b=*/false);
}

__device__ __forceinline__ float sigmoidf_(float x) {
  return 1.0f / (1.0f + expf(-x));
}

// ---------------------------------------------------------------------------
// Kernel 1: inputs = relu(concat(s,o) @ Wc^T + bc); also zero c, write newrep.
// ---------------------------------------------------------------------------
__global__ void combine_kernel(const float* __restrict__ s,
                               const float* __restrict__ o,
                               const float* __restrict__ Wc,
                               const float* __restrict__ bc,
                               float* __restrict__ h_ws,
                               float* __restrict__ c_ws,
                               float* __restrict__ newrep_out) {
  int gid = blockIdx.x * blockDim.x + threadIdx.x;
  if (gid >= B_ * H_) return;
  int b = gid / H_, h = gid % H_;
  const float4* sr = (const float4*)(s + (size_t)b * H_);
  const float4* orr = (const float4*)(o + (size_t)b * H_);
  const float4* w0 = (const float4*)(Wc + (size_t)h * K2_);
  const float4* w1 = w0 + (H_ / 4);
  float acc = bc[h];
  for (int j = 0; j < H_ / 4; ++j) {
    float4 a = sr[j], w = w0[j];
    acc += a.x * w.x + a.y * w.y + a.z * w.z + a.w * w.w;
  }
  for (int j = 0; j < H_ / 4; ++j) {
    float4 a = orr[j], w = w1[j];
    acc += a.x * w.x + a.y * w.y + a.z * w.z + a.w * w.w;
  }
  float r = acc > 0.0f ? acc : 0.0f;
  h_ws[gid] = r;
  c_ws[gid] = 0.0f;
  newrep_out[gid] = r;
}

// ---------------------------------------------------------------------------
// Kernel 2: xg[m,g] = E[tok[m]] @ W_ih^T + b_ih + b_hh   (M=1024, N=2048, K=512)
// Block: 8 waves, tile M=16 x N=128. Embedding rows async-gathered into LDS.
// ---------------------------------------------------------------------------
__global__ void xg_kernel(const int* __restrict__ targets,
                          const float* __restrict__ E,
                          const float* __restrict__ W_ih,
                          const float* __restrict__ b_ih,
                          const float* __restrict__ b_hh,
                          float* __restrict__ xg) {
  __shared__ float As[16 * D_];  // 32 KB A tile (gathered embeddings)
  const int m0 = blockIdx.y * 16;
  const int tid = threadIdx.x;

  for (int idx = tid; idx < 16 * (D_ / 4); idx += blockDim.x) {
    int r = idx / (D_ / 4), c4 = idx % (D_ / 4);
    int m = m0 + r;
    int bb = m / T_, tt = m % T_;
    int tok = (tt == 0) ? 1 : targets[bb * T_ + tt - 1];  // BOS=1, teacher forcing
    async_copy16(E + (size_t)tok * D_ + c4 * 4, As + idx * 4);
  }
  ASYNC_WAIT(0);
  __syncthreads();

  const int wave = tid >> 5, lane = tid & 31;
  const int hl = lane >> 4, col = lane & 15;
  const int n0 = blockIdx.x * 128 + wave * 16;
  const int ncol = n0 + col;
  const float* brow = W_ih + (size_t)ncol * D_;

  v8f acc = {};
  for (int k = 0; k < D_; k += 4) {
    v2f a = *(const v2f*)(As + col * D_ + k + 2 * hl);
    v2f bfr = *(const v2f*)(brow + k + 2 * hl);
    acc = wmma_f32_k4(a, bfr, acc);
  }
  float bias = b_ih[ncol] + b_hh[ncol];
  for (int j = 0; j < 8; ++j) {
    int m = m0 + j + 8 * hl;
    xg[(size_t)m * G_ + ncol] = acc[j] + bias;
  }
}

// ---------------------------------------------------------------------------
// Kernel 3 (x64, stream-serialized): one fused LSTM step.
// 4 blocks x 8 waves = 32 waves; wave w owns hh tile [hh0, hh0+16) and computes
// all four gate tiles (i,f,g,o) + the cell update in-register.
// ---------------------------------------------------------------------------
__global__ void lstm_step_kernel(const float* __restrict__ W_hh,
                                 const float* __restrict__ xg,
                                 const float* __restrict__ h_in,
                                 float* __restrict__ h_out,
                                 float* __restrict__ c_ws,
                                 float* __restrict__ hs_ws,
                                 int t) {
  __shared__ float Hs[16 * H_];  // 32 KB: full h_prev [16][512]
  const int tid = threadIdx.x;
  for (int idx = tid; idx < 16 * (H_ / 4); idx += blockDim.x)
    async_copy16(h_in + idx * 4, Hs + idx * 4);
  ASYNC_WAIT(0);
  __syncthreads();

  const int wave = tid >> 5, lane = tid & 31;
  const int hl = lane >> 4, col = lane & 15;
  const int hh0 = (blockIdx.x * 8 + wave) * 16;
  const int n = hh0 + col;

  const float* bi = W_hh + (size_t)(0 * H_ + n) * H_;
  const float* bf = W_hh + (size_t)(1 * H_ + n) * H_;
  const float* bg = W_hh + (size_t)(2 * H_ + n) * H_;
  const float* bo = W_hh + (size_t)(3 * H_ + n) * H_;

  v8f ai = {}, af = {}, ag = {}, ao = {};
  for (int k = 0; k < H_; k += 4) {
    v2f a = *(const v2f*)(Hs + col * H_ + k + 2 * hl);
    ai = wmma_f32_k4(a, *(const v2f*)(bi + k + 2 * hl), ai);
    af = wmma_f32_k4(a, *(const v2f*)(bf + k + 2 * hl), af);
    ag = wmma_f32_k4(a, *(const v2f*)(bg + k + 2 * hl), ag);
    ao = wmma_f32_k4(a, *(const v2f*)(bo + k + 2 * hl), ao);
  }

  for (int j = 0; j < 8; ++j) {
    int m = j + 8 * hl;  // batch index
    const float* xr = xg + (size_t)(m * T_ + t) * G_;
    float gi = ai[j] + xr[n];
    float gf = af[j] + xr[H_ + n];
    float gg = ag[j] + xr[2 * H_ + n];
    float go = ao[j] + xr[3 * H_ + n];
    float c_old = c_ws[m * H_ + n];
    float cn = sigmoidf_(gf) * c_old + sigmoidf_(gi) * tanhf(gg);
    float hn = sigmoidf_(go) * tanhf(cn);
    c_ws[m * H_ + n] = cn;
    h_out[m * H_ + n] = hn;
    hs_ws[(size_t)(m * T_ + t) * H_ + n] = hn;
  }
}

// ---------------------------------------------------------------------------
// Kernel 4: logits = hs @ Wo^T + bo   (M=1024, N=50257, K=512) — HBM-bound.
// Block: 8 waves, tile M=16 x N=128. Per-wave Wo chunks are double-buffered in
// LDS via async copies; ASYNCcnt (<=8) gates the software pipeline. No block
// barrier needed in the K loop since each wave's chunk is private.
// ---------------------------------------------------------------------------

// Stage Wo[n0..n0+15][k0..k0+63] (4 KB) into a per-wave LDS chunk. Out-of-range
// rows are clamped to V-1: output element (m,n) only depends on B column n, and
// columns >= V are never stored, so clamped data is harmless.
__device__ __forceinline__ void stage_b_chunk(const float* __restrict__ Wo,
                                              int n0, int k0, float* bs, int lane) {
  for (int it = 0; it < 8; ++it) {
    int idx = it * 32 + lane;  // index into 16x16 float4 grid
    int r = idx >> 4, c4 = idx & 15;
    int nn = n0 + r;
    if (nn > V_ - 1) nn = V_ - 1;
    async_copy16(Wo + (size_t)nn * H_ + k0 + c4 * 4, bs + idx * 4);
  }
}

__global__ void logits_kernel(const float* __restrict__ hs,
                              const float* __restrict__ Wo,
                              const float* __restrict__ bo,
                              float* __restrict__ out) {
  __shared__ float As[16 * H_];          // 32 KB hs tile (block-shared)
  __shared__ float Bs[8][2][16 * 64];    // 8 waves x 2 x 4 KB Wo chunks
  const int tid = threadIdx.x;
  const int m0 = blockIdx.y * 16;
  const int wave = tid >> 5, lane = tid & 31;
  const int hl = lane >> 4, col = lane & 15;
  const int n0 = blockIdx.x * 128 + wave * 16;

  for (int idx = tid; idx < 16 * (H_ / 4); idx += blockDim.x)
    async_copy16(hs + (size_t)m0 * H_ + idx * 4, As + idx * 4);
  stage_b_chunk(Wo, n0, 0, &Bs[wave][0][0], lane);
  ASYNC_WAIT(0);
  __syncthreads();

  v8f acc = {};
  int buf = 0;
  for (int k0 = 0; k0 < H_; k0 += 64, buf ^= 1) {
    if (k0 + 64 < H_) {
      SCHED_FENCE();  // keep next-chunk issue behind WMMAs reading buf^1
      stage_b_chunk(Wo, n0, k0 + 64, &Bs[wave][buf ^ 1][0], lane);
      ASYNC_WAIT(8);  // in-order completion: current chunk's 8 copies landed
    } else {
      ASYNC_WAIT(0);
    }
    const float* bsr = &Bs[wave][buf][0];
    for (int k = 0; k < 64; k += 4) {
      v2f a = *(const v2f*)(As + col * H_ + k0 + k + 2 * hl);
      v2f bfr = *(const v2f*)(bsr + col * 64 + k + 2 * hl);
      acc = wmma_f32_k4(a, bfr, acc);
    }
  }

  int nn = n0 + col;
  if (nn < V_) {
    float bias = bo[nn];
    for (int j = 0; j < 8; ++j) {
      int m = m0 + j + 8 * hl;
      out[(size_t)m * V_ + nn] = acc[j] + bias;
    }
  }
}

// ---------------------------------------------------------------------------
// Kernel 5: predictions[m] = argmax_n logits[m, n] (first occurrence).
// ---------------------------------------------------------------------------
__global__ void argmax_kernel(const float* __restrict__ logits,
                              float* __restrict__ pred) {
  __shared__ float sval[256];
  __shared__ int sidx[256];
  const int row = blockIdx.x;
  const float* r = logits + (size_t)row * V_;
  const int tid = threadIdx.x;
  float best = -INFINITY;
  int bidx = V_;
  for (int n = tid; n < V_; n += 256) {
    float v = r[n];
    if (v > best) { best = v; bidx = n; }
  }
  sval[tid] = best;
  sidx[tid] = bidx;
  __syncthreads();
  for (int s2 = 128; s2 > 0; s2 >>= 1) {
    if (tid < s2) {
      float v2 = sval[tid + s2];
      int i2 = sidx[tid + s2];
      if (v2 > sval[tid] || (v2 == sval[tid] && i2 < sidx[tid])) {
        sval[tid] = v2;
        sidx[tid] = i2;
      }
    }
    __syncthreads();
  }
  if (tid == 0) pred[row] = (float)sidx[0];
}

// ---------------------------------------------------------------------------
extern "C" void kernel_launch(void* const* d_in, const int* in_sizes, int n_in,
                              void* d_out, int out_size, void* d_ws, size_t ws_size,
                              hipStream_t stream) {
  const float* s   = (const float*)d_in[0];
  const float* o   = (const float*)d_in[1];
  const int*   tg  = (const int*)d_in[2];
  const float* Wc  = (const float*)d_in[3];
  const float* bc  = (const float*)d_in[4];
  const float* E   = (const float*)d_in[5];
  const float* Wih = (const float*)d_in[6];
  const float* Whh = (const float*)d_in[7];
  const float* bih = (const float*)d_in[8];
  const float* bhh = (const float*)d_in[9];
  const float* Wo  = (const float*)d_in[10];
  const float* bov = (const float*)d_in[11];

  float* out = (float*)d_out;
  float* logits = out;                              // [B*T*V]
  float* pred   = out + (size_t)B_ * T_ * V_;       // [B*T]
  float* newrep = pred + B_ * T_;                   // [B*H]

  // Workspace layout (floats): ~10.6 MB total
  float* ws = (float*)d_ws;
  float* h0 = ws;                                   // 8192
  float* h1 = h0 + B_ * H_;                         // 8192
  float* c  = h1 + B_ * H_;                         // 8192
  float* hsb = c + B_ * H_;                         // 524288
  float* xg = hsb + (size_t)B_ * T_ * H_;           // 2097152

  combine_kernel<<<(B_ * H_ + 255) / 256, 256, 0, stream>>>(s, o, Wc, bc, h0, c, newrep);

  dim3 g2(G_ / 128, (B_ * T_) / 16);
  xg_kernel<<<g2, 256, 0, stream>>>(tg, E, Wih, bih, bhh, xg);

  for (int t = 0; t < T_; ++t) {
    float* hin  = (t & 1) ? h1 : h0;
    float* hout = (t & 1) ? h0 : h1;
    lstm_step_kernel<<<4, 256, 0, stream>>>(Whh, xg, hin, hout, c, hsb, t);
  }

  dim3 g4((V_ + 127) / 128, (B_ * T_) / 16);
  logits_kernel<<<g4, 256, 0, stream>>>(hsb, Wo, bov, logits);

  argmax_kernel<<<B_ * T_, 256, 0, stream>>>(logits, pred);
}